// TopKRouter_37391985279401
// MI455X (gfx1250) — compile-verified
//
#include <hip/hip_runtime.h>

typedef float v2f __attribute__((ext_vector_type(2)));
typedef float v4f __attribute__((ext_vector_type(4)));
typedef float v8f __attribute__((ext_vector_type(8)));

#define D_DIM 2048
#define E_DIM 64
#define TOK_PER_BLOCK 16
#define WAVES_PER_BLOCK 8
#define K_CHUNK (D_DIM / WAVES_PER_BLOCK)    // 256 K-elements per wave
#define TILE_ELEMS (TOK_PER_BLOCK * E_DIM)   // 1024 floats = 4 KB per partial tile

// One block = 16 tokens. Each of the 8 waves computes the 16x64 partial-logit
// tile over its private K-chunk via V_WMMA_F32_16X16X4_F32, spills to its own
// LDS slice, then the block does a fixed-order (deterministic) cross-wave
// reduction and a 16-thread top-2 + softmax epilogue.
__global__ __launch_bounds__(256) void topk_router_kernel(
    const float* __restrict__ x,      // (n_tokens, 2048) row-major
    const float* __restrict__ gw,     // (64, 2048) row-major
    int*   __restrict__ idx_out,      // (n_tokens, 2)
    float* __restrict__ w_out)        // (n_tokens, 2)
{
    __shared__ float smem[WAVES_PER_BLOCK * TILE_ELEMS];   // 32 KB

    const int tid    = threadIdx.x;
    const int lane   = tid & 31;
    const int wave   = tid >> 5;
    const int tile_m = blockIdx.x * TOK_PER_BLOCK;

    const int half = lane >> 4;   // 0: K pair {0,1}, 1: K pair {2,3}
    const int l16  = lane & 15;
    const int kbase = wave * K_CHUNK;

    // A fragment: lane (half,l16) loads x[tile_m + l16][kbase + k0 + 2*half + {0,1}]
    const float* a_base = x  + (size_t)(tile_m + l16) * D_DIM + kbase + 2 * half;
    // B fragment: lane (half,l16) loads gw[etile*16 + l16][kbase + k0 + 2*half + {0,1}]
    const float* b_base = gw + (size_t)l16 * D_DIM + kbase + 2 * half;

    v8f acc0 = {}, acc1 = {}, acc2 = {}, acc3 = {};

#pragma unroll 4
    for (int k0 = 0; k0 < K_CHUNK; k0 += 4) {
        v2f a  = *(const v2f*)(a_base + k0);
        v2f b0 = *(const v2f*)(b_base + k0);
        v2f b1 = *(const v2f*)(b_base + k0 + 16 * D_DIM);
        v2f b2 = *(const v2f*)(b_base + k0 + 32 * D_DIM);
        v2f b3 = *(const v2f*)(b_base + k0 + 48 * D_DIM);
        // 8 args: (neg_a, A, neg_b, B, c_mod, C, reuse_a, reuse_b)
        acc0 = __builtin_amdgcn_wmma_f32_16x16x4_f32(false, a, false, b0, (short)0, acc0, false, false);
        acc1 = __builtin_amdgcn_wmma_f32_16x16x4_f32(false, a, false, b1, (short)0, acc1, false, false);
        acc2 = __builtin_amdgcn_wmma_f32_16x16x4_f32(false, a, false, b2, (short)0, acc2, false, false);
        acc3 = __builtin_amdgcn_wmma_f32_16x16x4_f32(false, a, false, b3, (short)0, acc3, false, false);
    }

    // Spill this wave's partial tile to its own LDS slice: slice[token*64 + expert].
    // C/D layout: VGPR r, lanes 0-15 -> M=r ; lanes 16-31 -> M=8+r ; N = l16.
    float* my = smem + wave * TILE_ELEMS;
    const int trow = half * 8;
#pragma unroll
    for (int r = 0; r < 8; ++r) {
        my[(trow + r) * E_DIM +  0 + l16] = acc0[r];
        my[(trow + r) * E_DIM + 16 + l16] = acc1[r];
        my[(trow + r) * E_DIM + 32 + l16] = acc2[r];
        my[(trow + r) * E_DIM + 48 + l16] = acc3[r];
    }

    __syncthreads();

    // Deterministic cross-wave reduction: thread tid owns elements [4*tid, 4*tid+4)
    // of the 1024-element tile; sum the 8 slices in fixed order, write into slice 0.
    {
        v4f sum = *(const v4f*)(smem + 4 * tid);
#pragma unroll
        for (int s = 1; s < WAVES_PER_BLOCK; ++s)
            sum += *(const v4f*)(smem + s * TILE_ELEMS + 4 * tid);
        *(v4f*)(smem + 4 * tid) = sum;
    }

    __syncthreads();

    // 16 threads each reduce one token: serial top-2 over 64 logits + softmax.
    if (tid < TOK_PER_BLOCK) {
        const float* row = smem + tid * E_DIM;
        float b0v = -3.402823466e38f, b1v = -3.402823466e38f;
        int b0i = 0, b1i = 0;
#pragma unroll 8
        for (int e = 0; e < E_DIM; ++e) {
            float v = row[e];
            if (v > b0v) { b1v = b0v; b1i = b0i; b0v = v; b0i = e; }
            else if (v > b1v) { b1v = v; b1i = e; }
        }
        // softmax over the two selected logits (b0v >= b1v, numerically safe)
        float e1 = __expf(b1v - b0v);
        float inv = 1.0f / (1.0f + e1);
        int g = tile_m + tid;
        idx_out[2 * g + 0] = b0i;
        idx_out[2 * g + 1] = b1i;
        w_out[2 * g + 0] = inv;
        w_out[2 * g + 1] = e1 * inv;
    }
}

extern "C" void kernel_launch(void* const* d_in, const int* in_sizes, int n_in,
                              void* d_out, int out_size, void* d_ws, size_t ws_size,
                              hipStream_t stream) {
    const float* x  = (const float*)d_in[0];
    const float* gw = (const float*)d_in[1];
    // top_k (d_in[2]) is fixed at 2 per the reference.

    const int n_tokens = in_sizes[0] / D_DIM;              // 16384
    int*   idx_out = (int*)d_out;                          // indices (n_tokens, 2) int32
    float* w_out   = (float*)d_out + (size_t)n_tokens * 2; // weights (n_tokens, 2) f32

    const int blocks = n_tokens / TOK_PER_BLOCK;           // 1024 blocks x 8 waves = 8192 waves

    topk_router_kernel<<<blocks, WAVES_PER_BLOCK * 32, 0, stream>>>(x, gw, idx_out, w_out);
}